// SVD_69398081568849
// MI455X (gfx1250) — compile-verified
//
#include <hip/hip_runtime.h>

// Weighted rank-1 Procrustes (radar SVD) for MI455X / gfx1250.
//
// Reference collapses to 4 weighted moments per batch + closed-form R,t
// (H = h*q*q^T is rank-1 symmetric, q=(1,RES,0)): memory-bound streaming
// reduction (112 MB @ 23.3 TB/s ~ 4.8 us floor). Bulk accumulation is f32
// VALU FMA on non-temporal loads; the cross-thread reduction runs through
// v_wmma_f32_16x16x4_f32 (f32 => bit-exact) with an all-ones B matrix to
// row-sum 64 partials per moment.

#define RESF      0.2592f
#define WIDTHF    640.0f
#define NPTS      2048
#define TPB       256

typedef __attribute__((ext_vector_type(2))) float v2f;
typedef __attribute__((ext_vector_type(8))) float v8f;

__global__ __launch_bounds__(TPB) void procrustes_rank1_kernel(
    const float* __restrict__ kp,    // keypoint_coords (B*2, N, 2)
    const float* __restrict__ tgt,   // tgt_coords      (B,   N, 2)
    const float* __restrict__ w,     // weights         (B,   1, N)
    float* __restrict__ out)         // (B, 3, 4) row-major
{
    const int b   = blockIdx.x;
    const int tid = threadIdx.x;
    const float cart_min = (WIDTHF * 0.5f - 0.5f) * RESF;   // 82.8144

    __shared__ float P[4][TPB];   // per-thread partial moments

    // Only c[...,1] is used by _to_radar; load coord pairs as 8B vectors.
    const v2f* kp2  = (const v2f*)kp;
    const v2f* tgt2 = (const v2f*)tgt;
    const long kbase = (long)(2 * b) * NPTS;   // keypoint_coords[::2]
    const long tbase = (long)b * NPTS;

    // ---- Stage A: streaming accumulation (HBM-bound, non-temporal) ----
    float sw = 0.f, ssx = 0.f, stx = 0.f, sts = 0.f;
#pragma unroll
    for (int i = 0; i < NPTS / TPB; ++i) {
        const int n = tid + i * TPB;
        const float wn = __builtin_nontemporal_load(&w[tbase + n]);
        const v2f  kc  = __builtin_nontemporal_load(&kp2[kbase + n]);
        const v2f  tc  = __builtin_nontemporal_load(&tgt2[tbase + n]);
        const float xs = __builtin_fmaf(-RESF, kc.y, cart_min); // radar x (src)
        const float xt = __builtin_fmaf(-RESF, tc.y, cart_min); // radar x (tgt)
        sw  += wn;
        ssx  = __builtin_fmaf(wn, xs, ssx);
        stx  = __builtin_fmaf(wn, xt, stx);
        sts  = __builtin_fmaf(wn * xt, xs, sts);
    }
    P[0][tid] = sw;  P[1][tid] = ssx;  P[2][tid] = stx;  P[3][tid] = sts;
    __syncthreads();

    // ---- Stage B1: fold 256 -> 64 partials per moment (all 256 threads) ----
    {
        const int j = tid >> 6, i = tid & 63;
        P[j][i] = P[j][i] + P[j][i + 64] + P[j][i + 128] + P[j][i + 192];
    }
    __syncthreads();

    // ---- Stage B2: WMMA row-sum reduction, 64 -> 1 per moment ----
    // A[m][k] = P[j][4m+k] (16x4 f32 A-fragment layout: lanes 0-15 hold
    // K=0/1 of row m=lane, lanes 16-31 hold K=2/3 of row m=lane-16).
    // B = ones => D[m][n] = row-sum(m), identical across n. Executed by all
    // waves redundantly so EXEC is all-ones (WMMA requirement).
    const int lane = tid & 31;
    const int m    = lane & 15;
    const int k0   = (lane < 16) ? 0 : 2;
    const v2f bones = {1.0f, 1.0f};
    float S[4];
#pragma unroll
    for (int j = 0; j < 4; ++j) {
        v2f a;
        a.x = P[j][4 * m + k0];
        a.y = P[j][4 * m + k0 + 1];
        v8f c = {};
        c = __builtin_amdgcn_wmma_f32_16x16x4_f32(
                false, a, false, bones, (short)0, c, false, false);
        // lanes<16 carry rows 0..7, lanes>=16 carry rows 8..15
        float rs = c[0] + c[1] + c[2] + c[3] + c[4] + c[5] + c[6] + c[7];
        S[j] = rs + __shfl_xor(rs, 16, 32);   // total over all 16 rows
    }

    // ---- Stage C: closed-form SVD epilogue (rank-1 symmetric H) ----
    if (tid == 0) {
        const float Sw = S[0], Ss = S[1], St = S[2], Sts = S[3];
        const float cs = Ss / Sw;            // src centroid (radar-x)
        const float ct = St / Sw;            // tgt centroid (radar-x)
        const float h  = Sts / Sw - ct * cs; // H = h * q q^T, q=(1,RES,0)
        const float s  = (h < 0.f) ? -1.f : 1.f;   // det(U)*det(Vh)
        const float r  = RESF;
        const float L2 = 1.f + r * r;
        // R = s*qq^T/L2 + pp^T/L2 + s*zz^T  (p = (-r,1,0)); R symmetric.
        const float R00 = (s + r * r) / L2;
        const float R01 = (r * (s - 1.f)) / L2;
        const float R11 = (s * r * r + 1.f) / L2;
        const float R22 = s;
        // centroids in 3-D: cen = x*q + (0, -cart_min, 0)
        const float scx = cs, scy = __builtin_fmaf(r, cs, -cart_min);
        const float tcx = ct, tcy = __builtin_fmaf(r, ct, -cart_min);
        // t = tgt_cen - R * src_cen   (z components vanish)
        const float t0 = tcx - (R00 * scx + R01 * scy);
        const float t1 = tcy - (R01 * scx + R11 * scy);
        float* o = out + (long)b * 12;
        o[0] = R00; o[1] = R01; o[2]  = 0.f; o[3]  = t0;
        o[4] = R01; o[5] = R11; o[6]  = 0.f; o[7]  = t1;
        o[8] = 0.f; o[9] = 0.f; o[10] = R22; o[11] = 0.f;
    }
}

extern "C" void kernel_launch(void* const* d_in, const int* in_sizes, int n_in,
                              void* d_out, int out_size, void* d_ws, size_t ws_size,
                              hipStream_t stream) {
    const float* kp  = (const float*)d_in[0];   // (B*2, N, 2) f32
    const float* tgt = (const float*)d_in[1];   // (B,   N, 2) f32
    const float* w   = (const float*)d_in[2];   // (B,   1, N) f32
    float* out = (float*)d_out;                 // (B, 3, 4) f32

    const int B = in_sizes[2] / NPTS;           // weights has B*N elements
    procrustes_rank1_kernel<<<dim3(B), dim3(TPB), 0, stream>>>(kp, tgt, w, out);
}